// TemporalDiseaseGNN_89275190215325
// MI455X (gfx1250) — compile-verified
//
#include <hip/hip_runtime.h>
#include <hip/hip_bf16.h>

// CDNA5 fp32 WMMA fragment types
typedef float v2f __attribute__((ext_vector_type(2)));
typedef float v8f __attribute__((ext_vector_type(8)));

#define HDIM 64
#define CIN_DIM 3
#define TSTEPS 6

// ---------------------------------------------------------------- utilities

__global__ void fill_kernel(float* __restrict__ p, float v, int n) {
    int i = blockIdx.x * blockDim.x + threadIdx.x;
    if (i < n) p[i] = v;
}

__device__ __forceinline__ float sigmoidf_(float x) {
    return 1.0f / (1.0f + expf(-x));
}

// ------------------------------------------------------------ degree / norm

__global__ void deg_kernel(const int* __restrict__ dst, float* __restrict__ deg, int E) {
    int e = blockIdx.x * blockDim.x + threadIdx.x;
    if (e < E) {
        __hip_atomic_fetch_add(&deg[dst[e]], 1.0f,
                               __ATOMIC_RELAXED, __HIP_MEMORY_SCOPE_AGENT);
    }
}

__global__ void rsqrt_kernel(float* __restrict__ d, int n) {
    int i = blockIdx.x * blockDim.x + threadIdx.x;
    if (i < n) d[i] = rsqrtf(d[i]);
}

// --------------------------------------------------- GCN layer 1 X@W1 (K=3)

__global__ void xw1_kernel(const float* __restrict__ x,   // [N,3]
                           const float* __restrict__ W1,  // [3,64]
                           float* __restrict__ out, int N) {
    int idx = blockIdx.x * blockDim.x + threadIdx.x;
    if (idx >= N * HDIM) return;
    int n = idx >> 6, j = idx & 63;
    float acc = 0.f;
#pragma unroll
    for (int c = 0; c < CIN_DIM; ++c)
        acc += x[n * CIN_DIM + c] * W1[c * HDIM + j];
    out[idx] = acc;
}

// --------------------------------------------------------- edge aggregation
// agg[dst] += xw[src] * dinv[src]*dinv[dst]; 16 threads per edge (float4 each)

__global__ void agg_kernel(const int* __restrict__ src, const int* __restrict__ dst,
                           const float* __restrict__ dinv,
                           const float* __restrict__ xw,
                           float* __restrict__ agg, int E) {
    int tid = blockIdx.x * blockDim.x + threadIdx.x;
    if (tid >= E * 16) return;
    int e = tid >> 4;
    int j = (tid & 15) << 2;
    int s = src[e], d = dst[e];
    float norm = dinv[s] * dinv[d];
    const float4 v = *(const float4*)(xw + s * HDIM + j);
    float* base = agg + d * HDIM + j;
    __hip_atomic_fetch_add(base + 0, v.x * norm, __ATOMIC_RELAXED, __HIP_MEMORY_SCOPE_AGENT);
    __hip_atomic_fetch_add(base + 1, v.y * norm, __ATOMIC_RELAXED, __HIP_MEMORY_SCOPE_AGENT);
    __hip_atomic_fetch_add(base + 2, v.z * norm, __ATOMIC_RELAXED, __HIP_MEMORY_SCOPE_AGENT);
    __hip_atomic_fetch_add(base + 3, v.w * norm, __ATOMIC_RELAXED, __HIP_MEMORY_SCOPE_AGENT);
}

// ------------------------------------------ self-loop + bias + ReLU (in agg)

__global__ void gcn_finalize_kernel(const float* __restrict__ xw,
                                    const float* __restrict__ dinv,
                                    const float* __restrict__ b,
                                    float* __restrict__ agg, int N) {
    int idx = blockIdx.x * blockDim.x + threadIdx.x;
    if (idx >= N * HDIM) return;
    int n = idx >> 6, j = idx & 63;
    float di = dinv[n];
    float v = agg[idx] + di * di * xw[idx] + b[j];
    agg[idx] = v > 0.f ? v : 0.f;
}

// -------------------------------------------------------- fp32 WMMA GEMM
// C[M x P] = A[M x KDIM] @ B (+bias, optional ReLU)
// B element (k,p) lives at B[k*bsk + p*bsp]  (row-major: bsk=P,bsp=1;
//                                             transposed weight: bsk=1,bsp=KDIM)
// One wave computes 2 stacked 16x16 tiles. All B fragments (16) and all A
// fragments for both tiles (32) are preloaded into distinct registers BEFORE
// the WMMA chains, so the global_load_b64s clause up and s_wait_loadcnt
// pipelines down across the 32 fully-unrolled V_WMMA_F32_16X16X4_F32.
// Requires M % 32 == 0, P % 16 == 0 (holds: M=20000, P in {64,192}).

template <int KDIM>
__global__ void wmma_gemm_f32_kernel(const float* __restrict__ A,
                                     const float* __restrict__ B, int bsk, int bsp,
                                     const float* __restrict__ bias,
                                     float* __restrict__ C, int P,
                                     int MPairs, int Ptiles, int relu) {
    constexpr int KSTEPS = KDIM / 4;
    int wave = (blockIdx.x * blockDim.x + threadIdx.x) >> 5;
    int lane = threadIdx.x & 31;
    if (wave >= MPairs * Ptiles) return;          // wave-uniform exit
    int mp = wave / Ptiles;
    int pt = wave - mp * Ptiles;

    int lhalf = lane >> 4;                        // 0: lanes 0-15, 1: lanes 16-31
    int l15   = lane & 15;
    int ncol  = pt * 16 + l15;                    // C column for this lane
    int khalf = lhalf << 1;                       // K offset 0 or 2 within a 4-step

    // Preload all B fragments (4x16 f32 each): lanes 0-15 -> K=kk,kk+1 ;
    // lanes 16-31 -> K=kk+2,kk+3 (mirrors the A striping).
    v2f bfrag[KSTEPS];
#pragma unroll
    for (int s = 0; s < KSTEPS; ++s) {
        int kb = s * 4 + khalf;
        const float* bp = B + kb * bsk + ncol * bsp;
        bfrag[s].x = bp[0];
        bfrag[s].y = bp[bsk];
    }

    // Preload all A fragments for both M-tiles (distinct registers so loads
    // pipeline instead of serializing on one temp).
    const float* arow0 = A + (mp * 2 + 0) * 16 * KDIM + l15 * KDIM + khalf;
    const float* arow1 = arow0 + 16 * KDIM;
    v2f afrag0[KSTEPS], afrag1[KSTEPS];
#pragma unroll
    for (int s = 0; s < KSTEPS; ++s) {
        afrag0[s].x = arow0[s * 4];
        afrag0[s].y = arow0[s * 4 + 1];
    }
#pragma unroll
    for (int s = 0; s < KSTEPS; ++s) {
        afrag1[s].x = arow1[s * 4];
        afrag1[s].y = arow1[s * 4 + 1];
    }

    v8f c0 = {}, c1 = {};
#pragma unroll
    for (int s = 0; s < KSTEPS; ++s) {
        c0 = __builtin_amdgcn_wmma_f32_16x16x4_f32(
                false, afrag0[s], false, bfrag[s], (short)0, c0, false, false);
        c1 = __builtin_amdgcn_wmma_f32_16x16x4_f32(
                false, afrag1[s], false, bfrag[s], (short)0, c1, false, false);
    }

    float badd = bias ? bias[ncol] : 0.f;
    int mbase = mp * 32 + (lhalf << 3);           // VGPR v: M=v (lanes 0-15), M=v+8 (16-31)
#pragma unroll
    for (int v = 0; v < 8; ++v) {
        float val = c0[v] + badd;
        if (relu) val = val > 0.f ? val : 0.f;
        C[(mbase + v) * P + ncol] = val;
    }
#pragma unroll
    for (int v = 0; v < 8; ++v) {
        float val = c1[v] + badd;
        if (relu) val = val > 0.f ? val : 0.f;
        C[(mbase + 16 + v) * P + ncol] = val;
    }
}

// ----------------------------------------------------------------- GRU cell

__global__ void gru_gate_kernel(const float* __restrict__ gi,   // [N,192]
                                const float* __restrict__ gh,   // [N,192]
                                float* __restrict__ h, int N) {
    int idx = blockIdx.x * blockDim.x + threadIdx.x;
    if (idx >= N * HDIM) return;
    int n = idx >> 6, j = idx & 63;
    const float* gin = gi + n * 3 * HDIM;
    const float* ghn = gh + n * 3 * HDIM;
    float r  = sigmoidf_(gin[j] + ghn[j]);
    float zg = sigmoidf_(gin[HDIM + j] + ghn[HDIM + j]);
    float ng = tanhf(gin[2 * HDIM + j] + r * ghn[2 * HDIM + j]);
    float hv = h[idx];
    h[idx] = (1.f - zg) * ng + zg * hv;
}

// ------------------------------------------------------------------- head

__global__ void head_kernel(const float* __restrict__ hidden,  // [N,64] (post-ReLU)
                            const float* __restrict__ W4,      // [64,1]
                            const float* __restrict__ b4,
                            float* __restrict__ out, int N) {
    int n = blockIdx.x * blockDim.x + threadIdx.x;
    if (n >= N) return;
    float acc = b4[0];
#pragma unroll
    for (int j = 0; j < HDIM; ++j) acc += hidden[n * HDIM + j] * W4[j];
    out[n] = sigmoidf_(acc);
}

// ----------------------------------------------------------------- launcher

static inline int cdiv(int a, int b) { return (a + b - 1) / b; }

extern "C" void kernel_launch(void* const* d_in, const int* in_sizes, int n_in,
                              void* d_out, int out_size, void* d_ws, size_t ws_size,
                              hipStream_t stream) {
    const float* node_features = (const float*)d_in[0];   // [T,N,3]
    const int*   edges         = (const int*)  d_in[1];   // [T,2,E]
    const float* W1   = (const float*)d_in[2];
    const float* b1   = (const float*)d_in[3];
    const float* W2   = (const float*)d_in[4];
    const float* b2   = (const float*)d_in[5];
    const float* w_ih = (const float*)d_in[6];            // [192,64]
    const float* w_hh = (const float*)d_in[7];            // [192,64]
    const float* b_ih = (const float*)d_in[8];
    const float* b_hh = (const float*)d_in[9];
    const float* W3   = (const float*)d_in[10];
    const float* b3   = (const float*)d_in[11];
    const float* W4   = (const float*)d_in[12];
    const float* b4   = (const float*)d_in[13];
    float* out = (float*)d_out;

    const int N = in_sizes[0] / (TSTEPS * CIN_DIM);       // 20000
    const int E = in_sizes[1] / (TSTEPS * 2);             // 640000

    // workspace layout (floats)
    float* ws   = (float*)d_ws;
    float* dinv = ws;                      // N
    float* t0   = dinv + N;                // N*64  (xw scratch / hidden)
    float* t1   = t0 + (size_t)N * HDIM;   // N*64  (agg / z)
    float* hbuf = t1 + (size_t)N * HDIM;   // N*64  (GRU state)
    float* gi   = hbuf + (size_t)N * HDIM; // N*192
    float* gh   = gi + (size_t)N * 3 * HDIM; // N*192
    size_t need = ((size_t)N * (1 + 3 * HDIM + 6 * HDIM)) * sizeof(float);
    if (ws_size < need) return;

    const int TB = 256;
    const int NH = N * HDIM;
    const int MPairs = N / 32;                         // 625 (N % 32 == 0)

    // h0 = 0
    fill_kernel<<<cdiv(NH, TB), TB, 0, stream>>>(hbuf, 0.f, NH);

    for (int t = 0; t < TSTEPS; ++t) {
        const float* x_t = node_features + (size_t)t * N * CIN_DIM;
        const int* src = edges + ((size_t)t * 2 + 0) * E;
        const int* dst = edges + ((size_t)t * 2 + 1) * E;

        // deg = 1 + segment_sum(1, dst);  dinv = rsqrt(deg)
        fill_kernel<<<cdiv(N, TB), TB, 0, stream>>>(dinv, 1.f, N);
        deg_kernel<<<cdiv(E, TB), TB, 0, stream>>>(dst, dinv, E);
        rsqrt_kernel<<<cdiv(N, TB), TB, 0, stream>>>(dinv, N);

        // ---- GCN layer 1: xw = x@W1 ; aggregate ; +selfloop +b1 ; ReLU -> t1
        xw1_kernel<<<cdiv(NH, TB), TB, 0, stream>>>(x_t, W1, t0, N);
        fill_kernel<<<cdiv(NH, TB), TB, 0, stream>>>(t1, 0.f, NH);
        agg_kernel<<<cdiv(E * 16, TB), TB, 0, stream>>>(src, dst, dinv, t0, t1, E);
        gcn_finalize_kernel<<<cdiv(NH, TB), TB, 0, stream>>>(t0, dinv, b1, t1, N);

        // ---- GCN layer 2: xw = z1@W2 (WMMA) ; aggregate ; finalize -> t1 (=z2)
        {
            int Pt = HDIM / 16, waves = MPairs * Pt;
            wmma_gemm_f32_kernel<HDIM><<<cdiv(waves, 4), 128, 0, stream>>>(
                t1, W2, HDIM, 1, nullptr, t0, HDIM, MPairs, Pt, 0);
        }
        fill_kernel<<<cdiv(NH, TB), TB, 0, stream>>>(t1, 0.f, NH);
        agg_kernel<<<cdiv(E * 16, TB), TB, 0, stream>>>(src, dst, dinv, t0, t1, E);
        gcn_finalize_kernel<<<cdiv(NH, TB), TB, 0, stream>>>(t0, dinv, b2, t1, N);

        // ---- GRU: gi = z2@w_ih^T + b_ih ; gh = h@w_hh^T + b_hh  (WMMA)
        {
            int P = 3 * HDIM, Pt = P / 16, waves = MPairs * Pt;
            wmma_gemm_f32_kernel<HDIM><<<cdiv(waves, 4), 128, 0, stream>>>(
                t1, w_ih, 1, HDIM, b_ih, gi, P, MPairs, Pt, 0);
            wmma_gemm_f32_kernel<HDIM><<<cdiv(waves, 4), 128, 0, stream>>>(
                hbuf, w_hh, 1, HDIM, b_hh, gh, P, MPairs, Pt, 0);
        }
        gru_gate_kernel<<<cdiv(NH, TB), TB, 0, stream>>>(gi, gh, hbuf, N);
    }

    // head: hidden = relu(h@W3 + b3) (WMMA) ; pred = sigmoid(hidden@W4 + b4)
    {
        int Pt = HDIM / 16, waves = MPairs * Pt;
        wmma_gemm_f32_kernel<HDIM><<<cdiv(waves, 4), 128, 0, stream>>>(
            hbuf, W3, HDIM, 1, b3, t0, HDIM, MPairs, Pt, 1);
    }
    head_kernel<<<cdiv(N, TB), TB, 0, stream>>>(t0, W4, b4, out, N);
    (void)out_size; (void)n_in;
}